// FDALoss_63505386438847
// MI455X (gfx1250) — compile-verified
//
#include <hip/hip_runtime.h>
#include <hip/hip_bf16.h>

// Problem constants (from reference setup_inputs)
#define NB   2
#define HH   48
#define WW   48
#define CC   384
#define KK   16
#define LL   (HH * WW)          // 2304
#define ROWS (NB * LL)          // 4608
#define DINO_SHIFT 0.7f
#define EPS_NRM 1e-12f

// Block covers 128(s) x 128(l); 8 waves in a 4(s) x 2(l) grid;
// each wave computes a 2(s) x 4(l) grid of 16x16 WMMA tiles (32 x 64 region).
#define SBLK (LL / 128)             // 18
#define LBLK (LL / 128)             // 18
#define NBLOCKS (NB * SBLK * LBLK)  // 648

typedef __attribute__((ext_vector_type(2))) float v2f;
typedef __attribute__((ext_vector_type(8))) float v8f;

// ---------------------------------------------------------------------------
// Kernel 1: row-wise L2 normalize of features (one 128-thread block per row)
// ---------------------------------------------------------------------------
__global__ __launch_bounds__(128) void fda_normalize_kernel(
    const float* __restrict__ f, float* __restrict__ fn) {
  const int row = blockIdx.x;
  const float* src = f + (size_t)row * CC;
  float* dst = fn + (size_t)row * CC;

  float x0 = src[threadIdx.x];
  float x1 = src[threadIdx.x + 128];
  float x2 = src[threadIdx.x + 256];
  float ss = x0 * x0 + x1 * x1 + x2 * x2;

  for (int o = 16; o > 0; o >>= 1) ss += __shfl_down(ss, o, 32);

  __shared__ float wsum[4];
  const int lane = threadIdx.x & 31, wid = threadIdx.x >> 5;
  if (lane == 0) wsum[wid] = ss;
  __syncthreads();
  const float tot = wsum[0] + wsum[1] + wsum[2] + wsum[3];
  const float inv = 1.0f / fmaxf(sqrtf(tot), EPS_NRM);

  dst[threadIdx.x]       = x0 * inv;
  dst[threadIdx.x + 128] = x1 * inv;
  dst[threadIdx.x + 256] = x2 * inv;
}

// ---------------------------------------------------------------------------
// Kernel 2: P = exp(logp), H[s] = sum_c P[s,c]*logp[s,c]  (one thread per row)
// ---------------------------------------------------------------------------
__global__ __launch_bounds__(256) void fda_ph_kernel(
    const float* __restrict__ logp, float* __restrict__ P,
    float* __restrict__ H) {
  const int r = blockIdx.x * blockDim.x + threadIdx.x;
  if (r >= ROWS) return;
  const float* p = logp + (size_t)r * KK;
  float* Pr = P + (size_t)r * KK;
  float h = 0.f;
#pragma unroll
  for (int k = 0; k < KK; ++k) {
    const float lp = p[k];
    const float e = expf(lp);
    Pr[k] = e;
    h += e * lp;
  }
  H[r] = h;
}

// ---------------------------------------------------------------------------
// Kernel 3: main tile kernel, register-blocked 2x4 tiles per wave.
//   f_corr = fn_s @ fn_l^T - 0.7        (K-reduce over C=384, x4 f32 WMMA)
//   p_corr = H_s - P_s @ logp_l^T       (K-reduce over KK=16, x4 f32 WMMA)
// A(16x4,f32) and B(4x16,f32) share the lane layout:
//   lane t, vgpr v <-> row (t&15), k = v + 2*(t>>4)  => plain float2 loads.
// C/D layout: vgpr v, lane t -> row v + 8*(t>>4), col (t&15).
// ---------------------------------------------------------------------------
__global__ __launch_bounds__(256) void fda_tile_kernel(
    const float* __restrict__ fn, const float* __restrict__ P,
    const float* __restrict__ logp, const float* __restrict__ H,
    float* __restrict__ partial) {
  const int wid  = threadIdx.x >> 5;
  const int lane = threadIdx.x & 31;

  const int bid  = blockIdx.x;
  const int n    = bid / (SBLK * LBLK);
  const int rem  = bid % (SBLK * LBLK);
  const int sblk = rem / LBLK;
  const int lblk = rem % LBLK;

  const int s_base = sblk * 128 + (wid >> 1) * 32;  // 2 s-tiles from here
  const int l_base = lblk * 128 + (wid & 1) * 64;   // 4 l-tiles from here

  const int rI = lane & 15;
  const int h  = lane >> 4;

  // ---- feature Gram tiles: 2x4 accumulators, K over C=384 in steps of 4 ----
  const float* As[2];
  const float* Bs[4];
  As[0] = fn + ((size_t)n * LL + s_base + rI) * CC + 2 * h;
  As[1] = As[0] + (size_t)16 * CC;
#pragma unroll
  for (int j = 0; j < 4; ++j)
    Bs[j] = fn + ((size_t)n * LL + l_base + j * 16 + rI) * CC + 2 * h;

  v8f facc[2][4] = {};
#pragma unroll 2
  for (int k0 = 0; k0 < CC; k0 += 4) {
    v2f a[2], b[4];
#pragma unroll
    for (int i = 0; i < 2; ++i) a[i] = *(const v2f*)(As[i] + k0);
#pragma unroll
    for (int j = 0; j < 4; ++j) b[j] = *(const v2f*)(Bs[j] + k0);
#pragma unroll
    for (int i = 0; i < 2; ++i)
#pragma unroll
      for (int j = 0; j < 4; ++j)
        facc[i][j] = __builtin_amdgcn_wmma_f32_16x16x4_f32(
            false, a[i], false, b[j], (short)0, facc[i][j], false, false);
  }

  // ---- probability cross tiles: K over KK=16 ----
  const float* pAs[2];
  const float* pBs[4];
  pAs[0] = P + ((size_t)n * LL + s_base + rI) * KK + 2 * h;
  pAs[1] = pAs[0] + (size_t)16 * KK;
#pragma unroll
  for (int j = 0; j < 4; ++j)
    pBs[j] = logp + ((size_t)n * LL + l_base + j * 16 + rI) * KK + 2 * h;

  v8f pacc[2][4] = {};
#pragma unroll
  for (int k0 = 0; k0 < KK; k0 += 4) {
    v2f a[2], b[4];
#pragma unroll
    for (int i = 0; i < 2; ++i) a[i] = *(const v2f*)(pAs[i] + k0);
#pragma unroll
    for (int j = 0; j < 4; ++j) b[j] = *(const v2f*)(pBs[j] + k0);
#pragma unroll
    for (int i = 0; i < 2; ++i)
#pragma unroll
      for (int j = 0; j < 4; ++j)
        pacc[i][j] = __builtin_amdgcn_wmma_f32_16x16x4_f32(
            false, a[i], false, b[j], (short)0, pacc[i][j], false, false);
  }

  // ---- masked accumulation ----
  const float* Hn = H + (size_t)n * LL;
  float hv[2][8];
#pragma unroll
  for (int i = 0; i < 2; ++i)
#pragma unroll
    for (int v = 0; v < 8; ++v)
      hv[i][v] = Hn[s_base + i * 16 + v + 8 * h];

  float pos = 0.f, neg = 0.f, npos = 0.f, nneg = 0.f;
#pragma unroll
  for (int i = 0; i < 2; ++i)
#pragma unroll
    for (int j = 0; j < 4; ++j)
#pragma unroll
      for (int v = 0; v < 8; ++v) {
        const float pc = hv[i][v] - pacc[i][j][v];
        const float fc = facc[i][j][v] - DINO_SHIFT;
        pos  += fmaxf(fc, 0.f) * pc;
        neg  += fminf(fc, 0.f) * pc;
        npos += (fc > 0.f) ? 1.f : 0.f;
        nneg += (fc < 0.f) ? 1.f : 0.f;
      }

  // wave32 reduce
  for (int o = 16; o > 0; o >>= 1) {
    pos  += __shfl_down(pos,  o, 32);
    neg  += __shfl_down(neg,  o, 32);
    npos += __shfl_down(npos, o, 32);
    nneg += __shfl_down(nneg, o, 32);
  }

  __shared__ float sm[8][4];
  if (lane == 0) {
    sm[wid][0] = pos; sm[wid][1] = neg; sm[wid][2] = npos; sm[wid][3] = nneg;
  }
  __syncthreads();
  if (threadIdx.x == 0) {
    float a0 = 0.f, a1 = 0.f, a2 = 0.f, a3 = 0.f;
#pragma unroll
    for (int w = 0; w < 8; ++w) {
      a0 += sm[w][0]; a1 += sm[w][1]; a2 += sm[w][2]; a3 += sm[w][3];
    }
    float* out = partial + (size_t)bid * 4;
    out[0] = a0; out[1] = a1; out[2] = a2; out[3] = a3;
  }
}

// ---------------------------------------------------------------------------
// Kernel 4: deterministic final reduce of per-block partials + scalar combine
// ---------------------------------------------------------------------------
__global__ __launch_bounds__(256) void fda_reduce_kernel(
    const float* __restrict__ part, const float* __restrict__ pw,
    const float* __restrict__ nw, float* __restrict__ out) {
  float a0 = 0.f, a1 = 0.f, a2 = 0.f, a3 = 0.f;
  for (int i = threadIdx.x; i < NBLOCKS; i += 256) {
    a0 += part[i * 4 + 0];
    a1 += part[i * 4 + 1];
    a2 += part[i * 4 + 2];
    a3 += part[i * 4 + 3];
  }
  __shared__ float sm[4][256];
  sm[0][threadIdx.x] = a0; sm[1][threadIdx.x] = a1;
  sm[2][threadIdx.x] = a2; sm[3][threadIdx.x] = a3;
  __syncthreads();
  if (threadIdx.x == 0) {
    float s0 = 0.f, s1 = 0.f, s2 = 0.f, s3 = 0.f;
    for (int i = 0; i < 256; ++i) {
      s0 += sm[0][i]; s1 += sm[1][i]; s2 += sm[2][i]; s3 += sm[3][i];
    }
    out[0] = (s0 / s2) * pw[0] + (s1 / s3) * nw[0];  // COEF = 1.0
  }
}

// ---------------------------------------------------------------------------
extern "C" void kernel_launch(void* const* d_in, const int* in_sizes, int n_in,
                              void* d_out, int out_size, void* d_ws, size_t ws_size,
                              hipStream_t stream) {
  const float* f    = (const float*)d_in[0];  // [N,H,W,C] f32
  const float* logp = (const float*)d_in[1];  // [N,H,W,K] f32 (log-probs)
  const float* pw   = (const float*)d_in[2];  // [1]
  const float* nw   = (const float*)d_in[3];  // [1]
  float* out = (float*)d_out;                 // scalar f32

  // workspace layout
  float* ws      = (float*)d_ws;
  float* fn      = ws;                                  // ROWS*CC   = 1,769,472
  float* P       = fn + (size_t)ROWS * CC;              // ROWS*KK   = 73,728
  float* H       = P + (size_t)ROWS * KK;               // ROWS      = 4,608
  float* partial = H + ROWS;                            // NBLOCKS*4 = 2,592

  fda_normalize_kernel<<<ROWS, 128, 0, stream>>>(f, fn);
  fda_ph_kernel<<<(ROWS + 255) / 256, 256, 0, stream>>>(logp, P, H);
  fda_tile_kernel<<<NBLOCKS, 256, 0, stream>>>(fn, P, logp, H, partial);
  fda_reduce_kernel<<<1, 256, 0, stream>>>(partial, pw, nw, out);
}